// DenseDilatedKnnGraph_47347719471628
// MI455X (gfx1250) — compile-verified
//
#include <hip/hip_runtime.h>
#include <hip/hip_bf16.h>

// DenseDilatedKnnGraph for MI455X (gfx1250), fully fused:
//   stage 1: column squared-norms  -> d_ws
//   stage 2: per-16-row-stripe f32 WMMA Gram tiles + register-resident top-18
// Never materializes the (B,N,N) distance matrix (saves ~0.5 GB HBM traffic);
// x (4 MB) is L2-resident on a 192 MB L2, so the O(N^2 C / 16) re-reads of x
// are cache hits, not HBM traffic.

typedef __attribute__((ext_vector_type(2))) float v2f;
typedef __attribute__((ext_vector_type(8))) float v8f;

#define BB 4
#define CC 64
#define NN 4096
#define KEFF 18
#define KOUT 9

__global__ __launch_bounds__(256) void knn_norms_kernel(const float* __restrict__ x,
                                                        float* __restrict__ sq) {
    int idx = blockIdx.x * blockDim.x + threadIdx.x;   // b*NN + n
    if (idx >= BB * NN) return;
    int b = idx >> 12;
    int n = idx & (NN - 1);
    const float* xp = x + (size_t)b * CC * NN + n;
    float s = 0.0f;
#pragma unroll
    for (int c = 0; c < CC; ++c) {
        float v = xp[(size_t)c * NN];
        s = fmaf(v, v, s);
    }
    sq[idx] = s;
}

__global__ __launch_bounds__(32) void knn_topk_kernel(const float* __restrict__ x,
                                                      const float* __restrict__ sq,
                                                      int* __restrict__ out) {
    __shared__ float ldsD[16][17];   // padded: conflict-free row reads

    const int wg   = blockIdx.x;            // 0 .. B*(N/16)-1
    const int b    = wg >> 8;                // wg / (NN/16)
    const int m0   = (wg & 255) << 4;        // row-tile base
    const int lane = threadIdx.x;            // 0..31 (wave32)
    const int ml   = lane & 15;
    const int hi   = lane >> 4;              // 0 | 1
    const int kb_off = hi << 1;              // K sub-offset: 0 | 2 (f32 A/B layout)

    const float* xb = x + (size_t)b * CC * NN;

    // ---- A fragments for all 16 K-steps (16x4 f32 each), pre-scaled by -2 ----
    // Lane l holds M = l&15; VGPR0 = K = 4s + (hi?2:0), VGPR1 = K+1.
    v2f a_frag[16];
#pragma unroll
    for (int s = 0; s < 16; ++s) {
        int kb = 4 * s + kb_off;
        a_frag[s].x = -2.0f * xb[(size_t)kb * NN + (m0 + ml)];
        a_frag[s].y = -2.0f * xb[(size_t)(kb + 1) * NN + (m0 + ml)];
    }

    // ---- running sorted top-18 (ascending key), valid in lanes 0..15 ----
    float d_best[KEFF];
    int   i_best[KEFF];
#pragma unroll
    for (int j = 0; j < KEFF; ++j) { d_best[j] = 3.4e38f; i_best[j] = 0x7fffffff; }

    for (int n0 = 0; n0 < NN; n0 += 16) {
        // ---- 16 chained v_wmma_f32_16x16x4_f32: acc = -2 * A * B ----
        v8f acc = {};
#pragma unroll
        for (int s = 0; s < 16; ++s) {
            int kb = 4 * s + kb_off;
            v2f bfrag;
            bfrag.x = xb[(size_t)kb * NN + (n0 + ml)];
            bfrag.y = xb[(size_t)(kb + 1) * NN + (n0 + ml)];
            acc = __builtin_amdgcn_wmma_f32_16x16x4_f32(
                /*neg_a=*/false, a_frag[s], /*neg_b=*/false, bfrag,
                /*c_mod=*/(short)0, acc, /*reuse_a=*/false, /*reuse_b=*/false);
        }

        // key(m,n) = ||x_n||^2 - 2<x_m,x_n>  (row-constant ||x_m||^2 dropped:
        // it cannot change the per-row ranking)
        float sqn = sq[b * NN + n0 + ml];
        int rbase = hi << 3;                 // C/D layout: VGPR v -> row v + 8*hi
#pragma unroll
        for (int v = 0; v < 8; ++v) {
            ldsD[rbase + v][ml] = sqn + acc[v];
        }
        __syncthreads();

        // ---- lanes 0..15: insert this tile's 16 candidates for own row ----
        if (lane < 16) {
#pragma unroll
            for (int col = 0; col < 16; ++col) {
                float d = ldsD[lane][col];
                // strict < : on ties the earlier (smaller) index wins, matching
                // jax.lax.top_k tie-breaking (columns scanned in ascending order)
                if (d < d_best[KEFF - 1]) {
                    d_best[KEFF - 1] = d;
                    i_best[KEFF - 1] = n0 + col;
#pragma unroll
                    for (int j = KEFF - 1; j > 0; --j) {
                        if (d_best[j] < d_best[j - 1]) {
                            float td = d_best[j]; d_best[j] = d_best[j - 1]; d_best[j - 1] = td;
                            int   ti = i_best[j]; i_best[j] = i_best[j - 1]; i_best[j - 1] = ti;
                        }
                    }
                }
            }
        }
        __syncthreads();
    }

    // ---- emit dilated edge index: (2, B, N, 9), e=0 -> nn_idx, e=1 -> center ----
    if (lane < 16) {
        int n = m0 + lane;
        int* o0 = out + ((size_t)b * NN + n) * KOUT;            // nn_idx[b,n,::2]
        int* o1 = out + ((size_t)(BB + b) * NN + n) * KOUT;     // center = n
#pragma unroll
        for (int kk = 0; kk < KOUT; ++kk) {
            o0[kk] = i_best[2 * kk];
            o1[kk] = n;
        }
    }
}

extern "C" void kernel_launch(void* const* d_in, const int* in_sizes, int n_in,
                              void* d_out, int out_size, void* d_ws, size_t ws_size,
                              hipStream_t stream) {
    const float* x  = (const float*)d_in[0];       // (4, 64, 4096, 1) f32
    float*       sq = (float*)d_ws;                // B*N f32 column norms (64 KB)
    int*         out = (int*)d_out;                // (2, 4, 4096, 9) int32

    (void)in_sizes; (void)n_in; (void)out_size; (void)ws_size;

    // stage 1: column norms
    knn_norms_kernel<<<(BB * NN + 255) / 256, 256, 0, stream>>>(x, sq);

    // stage 2: fused WMMA Gram + streaming top-18 + dilated emit
    knn_topk_kernel<<<BB * (NN / 16), 32, 0, stream>>>(x, sq, out);
}